// MeanAggregator_79714593014087
// MI455X (gfx1250) — compile-verified
//
#include <hip/hip_runtime.h>

typedef float v2f __attribute__((ext_vector_type(2)));
typedef float v4f __attribute__((ext_vector_type(4)));
typedef float v8f __attribute__((ext_vector_type(8)));

#define D_   128   // feature dim
#define S_   25    // neighbor samples
#define OUT_ 128   // output dim
#define TM_  16    // node-tile (rows per block)
#define KDIM 256   // 2*D concat dim
#define PITCH 260  // LDS row pitch in floats (2*D + 4 pad -> conflict-free column reads)

__global__ __launch_bounds__(256) void sage_fused_wmma(
    const float* __restrict__ self_feat,   // [N, D]
    const float* __restrict__ neigh_feat,  // [N, S, D]
    const float* __restrict__ w,           // [2D, OUT]
    float* __restrict__ out,               // [N, OUT]
    int N)
{
    __shared__ float cat[TM_ * PITCH];     // [16 x 256] concat tile (padded)

    const int n0  = blockIdx.x * TM_;
    const int tid = threadIdx.x;

    // ---- Phase 1: fused mean-aggregation + concat into LDS ----
    // 16 nodes x 32 float4-chunks = 512 slots; 256 threads take 2 each.
    for (int s = tid; s < TM_ * (D_ / 4); s += 256) {
        const int node = s >> 5;       // s / 32
        const int f4   = s & 31;       // float4 chunk within feature dim
        int gnode = n0 + node;
        if (gnode >= N) gnode = N - 1; // clamp (keeps EXEC uniform later)

        // Neighbor stream: read-once -> non-temporal (don't evict w from L2)
        const v4f* np = (const v4f*)neigh_feat
                      + (size_t)gnode * (S_ * (D_ / 4)) + f4;
        v4f acc = {0.f, 0.f, 0.f, 0.f};
        #pragma unroll
        for (int si = 0; si < S_; ++si) {
            v4f v = __builtin_nontemporal_load(&np[si * (D_ / 4)]);
            acc += v;
        }
        acc *= (1.0f / (float)S_);

        v4f sf = ((const v4f*)self_feat)[(size_t)gnode * (D_ / 4) + f4];

        float* row = &cat[node * PITCH];
        *(v4f*)(row + 4 * f4)      = sf;   // cat[:, 0:128]   = self
        *(v4f*)(row + D_ + 4 * f4) = acc;  // cat[:, 128:256] = mean(neigh)
    }
    __syncthreads();

    // ---- Phase 2: [16 x 256] @ [256 x 128] via V_WMMA_F32_16X16X4_F32 ----
    const int wave = tid >> 5;
    const int lane = tid & 31;
    const int half = lane >> 4;   // 0: K0/K1 rows, 1: K2/K3 rows
    const int m    = lane & 15;
    const int col  = wave * 16 + m;   // this wave's 16 output columns

    v8f acc = {};
    #pragma unroll 8
    for (int kb = 0; kb < KDIM; kb += 4) {
        // A fragment (16x4 fp32): lanes 0-15 hold {K=kb, kb+1} of row m,
        // lanes 16-31 hold {K=kb+2, kb+3}.
        const float* ar = &cat[m * PITCH + kb + 2 * half];
        v2f a; a.x = ar[0]; a.y = ar[1];
        // B fragment (4x16 fp32): VGPR0 = rows {kb, kb+2}, VGPR1 = rows {kb+1, kb+3}
        const float* br = &w[(size_t)(kb + 2 * half) * OUT_ + col];
        v2f b; b.x = br[0]; b.y = br[OUT_];

        acc = __builtin_amdgcn_wmma_f32_16x16x4_f32(
                  /*neg_a=*/false, a, /*neg_b=*/false, b,
                  /*c_mod=*/(short)0, acc, /*reuse_a=*/false, /*reuse_b=*/false);
    }

    // ---- Store 16x16 f32 C/D tile: VGPR i -> row (i + 8*half), col ----
    float* op = out + (size_t)n0 * OUT_;
    if (n0 + TM_ <= N) {
        // Full tile (always taken for N % 16 == 0): uniform, unguarded,
        // non-temporal stores — no per-element EXEC manipulation.
        #pragma unroll
        for (int i = 0; i < 8; ++i) {
            const int row = i + 8 * half;
            __builtin_nontemporal_store(acc[i], &op[row * OUT_ + col]);
        }
    } else {
        #pragma unroll
        for (int i = 0; i < 8; ++i) {
            const int row = i + 8 * half;
            if (n0 + row < N) op[row * OUT_ + col] = acc[i];
        }
    }
}

extern "C" void kernel_launch(void* const* d_in, const int* in_sizes, int n_in,
                              void* d_out, int out_size, void* d_ws, size_t ws_size,
                              hipStream_t stream) {
    // setup_inputs order: src, src_neg, dst, dst_neg, w
    const float* src     = (const float*)d_in[0];
    const float* src_neg = (const float*)d_in[1];
    const float* dst     = (const float*)d_in[2];
    const float* dst_neg = (const float*)d_in[3];
    const float* w       = (const float*)d_in[4];
    float*       out     = (float*)d_out;

    const int N = in_sizes[0] / D_;           // 50000
    dim3 grid((N + TM_ - 1) / TM_);           // 3125 tiles
    dim3 block(256);                          // 8 wave32

    // Output tuple order: (re_src, re_dst) concatenated flat.
    sage_fused_wmma<<<grid, block, 0, stream>>>(src, src_neg, w, out, N);
    sage_fused_wmma<<<grid, block, 0, stream>>>(dst, dst_neg, w, out + (size_t)N * OUT_, N);
}